// GraphAttention_70480413327458
// MI455X (gfx1250) — compile-verified
//
#include <hip/hip_runtime.h>

typedef float v2f __attribute__((ext_vector_type(2)));
typedef float v8f __attribute__((ext_vector_type(8)));

#define GRIDN    25
#define NUM_NODE 625          // 25*25
#define FIN      66
#define FOUT     64
#define KPAD     68           // 66 padded up to multiple of 4
#define HSTRIDE  65           // h row stride in LDS (floats) -> conflict-free column walks
#define NBATCH   512

// One workgroup per batch image. 256 threads = 8 waves (wave32).
// LDS: h(625x65 f32)=162.5KB + Wpad(64x68)=17.4KB + Wat 512B + a1/a2/sw 7.5KB  ~= 188KB < 320KB/WGP.
__launch_bounds__(256, 1)
__global__ void gat_fused_kernel(const float* __restrict__ x,
                                 const float* __restrict__ W_fc,
                                 const float* __restrict__ b_fc,
                                 const float* __restrict__ W_at,
                                 const float* __restrict__ b_at,
                                 float* __restrict__ out_vec,   // [512,625,64]
                                 float* __restrict__ out_w)     // [512,625,9]
{
    __shared__ float sH[NUM_NODE * HSTRIDE];
    __shared__ float sW[FOUT * KPAD];
    __shared__ float sWat[128];
    __shared__ float sA1[NUM_NODE];
    __shared__ float sA2[NUM_NODE];
    __shared__ float sSW[NUM_NODE];

    const int tid  = threadIdx.x;
    const int b    = blockIdx.x;
    const int lane = tid & 31;
    const int wave = tid >> 5;

    // ---- Phase 0: stage weights into LDS (zero-padded K) ----
    for (int i = tid; i < FOUT * KPAD; i += 256) {
        int n = i / KPAD;
        int k = i - n * KPAD;
        sW[i] = (k < FIN) ? W_fc[n * FIN + k] : 0.0f;
    }
    if (tid < 128) sWat[tid] = W_at[tid];
    __syncthreads();

    // ---- Phase 1: h = x @ W_fc^T + b_fc via V_WMMA_F32_16X16X4_F32 ----
    // A (16x4): lane L<16 -> row M=L, regs hold K={0,1}; lanes 16..31 -> K={2,3}
    // B (4x16): lane L<16 -> col N=L, regs hold K={0,1}; lanes 16..31 -> K={2,3}
    // D (16x16): reg v -> row M = v + 8*(lane>=16), col N = lane&15
    const int nm_lo  = lane & 15;
    const int khalf  = (lane >> 4) << 1;            // 0 or 2
    const float* xb  = x + (size_t)b * NUM_NODE * FIN;

    float bias[4];
#pragma unroll
    for (int nt = 0; nt < 4; ++nt) bias[nt] = b_fc[nt * 16 + nm_lo];

    for (int m = wave; m < 40; m += 8) {            // 40 M-tiles of 16 rows
        int  row   = m * 16 + nm_lo;
        bool valid = row < NUM_NODE;
        // Clamp to a valid row so ALL loads are unconditional (no exec branching);
        // invalid lanes are zeroed with selects below.
        const float* xrow = xb + (size_t)(valid ? row : (NUM_NODE - 1)) * FIN;

        // Prefetch next M-tile's row while this tile computes.
        {
            int mn = m + 8;
            if (mn < 40) {
                int rown = mn * 16 + nm_lo;
                rown = (rown < NUM_NODE) ? rown : (NUM_NODE - 1);
                __builtin_prefetch(xb + (size_t)rown * FIN, 0, 0);
            }
        }

        v8f acc[4];
#pragma unroll
        for (int nt = 0; nt < 4; ++nt)
#pragma unroll
            for (int j = 0; j < 8; ++j) acc[nt][j] = 0.0f;

        // Main k-steps 0..15: columns <= 63+2 < 66, always in-bounds.
#pragma unroll
        for (int ks = 0; ks < 16; ++ks) {
            int c0 = ks * 4 + khalf;
            float t0 = xrow[c0];                    // unconditional global_load_b64
            float t1 = xrow[c0 + 1];
            v2f a;
            a.x = valid ? t0 : 0.0f;                // v_cndmask, no branches
            a.y = valid ? t1 : 0.0f;
#pragma unroll
            for (int nt = 0; nt < 4; ++nt) {
                const float* wp = &sW[(nt * 16 + nm_lo) * KPAD + c0];
                v2f bf;
                bf.x = wp[0];
                bf.y = wp[1];
                acc[nt] = __builtin_amdgcn_wmma_f32_16x16x4_f32(
                    false, a, false, bf, (short)0, acc[nt], false, false);
            }
        }

        // Tail k-step (kbase=64): khalf==0 lanes read cols 64/65 (valid),
        // khalf==2 lanes would read cols 66/67 (pad) -> select zero. Load only
        // cols 64/65 so nothing ever reads past the end of x.
        {
            float t0 = xrow[64];
            float t1 = xrow[65];
            bool tv = valid && (khalf == 0);
            v2f a;
            a.x = tv ? t0 : 0.0f;
            a.y = tv ? t1 : 0.0f;
            int c0 = 64 + khalf;                    // sW is zero-padded to 68
#pragma unroll
            for (int nt = 0; nt < 4; ++nt) {
                const float* wp = &sW[(nt * 16 + nm_lo) * KPAD + c0];
                v2f bf;
                bf.x = wp[0];
                bf.y = wp[1];
                acc[nt] = __builtin_amdgcn_wmma_f32_16x16x4_f32(
                    false, a, false, bf, (short)0, acc[nt], false, false);
            }
        }

        // store D tiles (+bias) into LDS h
#pragma unroll
        for (int v = 0; v < 8; ++v) {
            int rowD = m * 16 + v + ((lane >> 4) << 3);
            if (rowD < NUM_NODE) {
#pragma unroll
                for (int nt = 0; nt < 4; ++nt)
                    sH[rowD * HSTRIDE + nt * 16 + nm_lo] = acc[nt][v] + bias[nt];
            }
        }
    }
    __syncthreads();

    // ---- Phase 2a: a1[n] = h[n]·W_at[:64], a2[n] = h[n]·W_at[64:] ----
    for (int n = tid; n < NUM_NODE; n += 256) {
        const float* hrow = &sH[n * HSTRIDE];       // stride 65 -> no bank conflicts
        float a1 = 0.f, a2 = 0.f;
#pragma unroll 8
        for (int f = 0; f < FOUT; ++f) {
            float hv = hrow[f];
            a1 += hv * sWat[f];
            a2 += hv * sWat[64 + f];
        }
        sA1[n] = a1;
        sA2[n] = a2;
    }
    __syncthreads();

    // ---- Phase 2b: leaky-relu + softmax over 9 mirrored neighbors ----
    const float bat = b_at[0];
    float* outw_b = out_w + (size_t)b * NUM_NODE * 9;
    for (int n = tid; n < NUM_NODE; n += 256) {
        int r = n / GRIDN, c = n - r * GRIDN;
        int radj = (r == 0) - (r == GRIDN - 1);
        int cadj = (c == 0) - (c == GRIDN - 1);
        float a1n = sA1[n];
        float s[9];
        float mx = -3.4e38f;
#pragma unroll
        for (int k = 0; k < 9; ++k) {
            int nr = r + (k / 3 - 1) + radj;
            int nc = c + (k % 3 - 1) + cadj;
            float v = a1n + sA2[nr * GRIDN + nc] + bat;
            v = (v > 0.f) ? v : 0.01f * v;          // leaky_relu(0.01)
            s[k] = v;
            mx = fmaxf(mx, v);
        }
        float sum = 0.f;
#pragma unroll
        for (int k = 0; k < 9; ++k) { s[k] = __expf(s[k] - mx); sum += s[k]; }
        float rs = 1.0f / sum;
        float sw = 0.f;
#pragma unroll
        for (int k = 0; k < 9; ++k) {
            float w = s[k] * rs;
            sw += w;
            outw_b[n * 9 + k] = w;
        }
        sSW[n] = sw;
    }
    __syncthreads();

    // ---- Phase 2c: attn_vector = h * sum(attn_weight), coalesced float4 ----
    float* outv_b = out_vec + (size_t)b * NUM_NODE * FOUT;
    for (int i = tid; i < NUM_NODE * (FOUT / 4); i += 256) {
        int n = i >> 4;
        int f = (i & 15) << 2;
        float sw = sSW[n];
        const float* hrow = &sH[n * HSTRIDE + f];
        float4 o;
        o.x = hrow[0] * sw;
        o.y = hrow[1] * sw;
        o.z = hrow[2] * sw;
        o.w = hrow[3] * sw;
        *(float4*)(&outv_b[(size_t)n * FOUT + f]) = o;
    }
}

extern "C" void kernel_launch(void* const* d_in, const int* in_sizes, int n_in,
                              void* d_out, int out_size, void* d_ws, size_t ws_size,
                              hipStream_t stream) {
    const float* x    = (const float*)d_in[0];   // [512,625,66]
    const float* W_fc = (const float*)d_in[1];   // [64,66]
    const float* b_fc = (const float*)d_in[2];   // [64]
    const float* W_at = (const float*)d_in[3];   // [1,128]
    const float* b_at = (const float*)d_in[4];   // [1]

    float* out_vec = (float*)d_out;                                   // 512*625*64
    float* out_w   = out_vec + (size_t)NBATCH * NUM_NODE * FOUT;      // 512*625*9

    gat_fused_kernel<<<NBATCH, 256, 0, stream>>>(x, W_fc, b_fc, W_at, b_at,
                                                 out_vec, out_w);
}